// GRAPH_VAE_V3_41188736369180
// MI455X (gfx1250) — compile-verified
//
#include <hip/hip_runtime.h>

#define NN 100000          // N_NODES
#define NE 1600000         // N_EDGES
#define NB 16              // batch B
#define NMAXN 4096
#define ROWS (NB * NMAXN)  // 65536
#define LATD 64
#define DEC_IN 66
#define DEC_H 128

typedef __attribute__((ext_vector_type(2))) float v2f;
typedef __attribute__((ext_vector_type(8))) float v8f;

static __device__ __forceinline__ long long gtid() {
  return (long long)blockIdx.x * blockDim.x + threadIdx.x;
}

// ---------------- utility ----------------
__global__ void k_fill(float* __restrict__ p, float v, long long n) {
  long long i = gtid();
  if (i < n) p[i] = v;
}

__global__ void k_deg(const int* __restrict__ dst, float* __restrict__ deg, int E) {
  long long e = gtid();
  if (e < E) atomicAdd(&deg[dst[e]], 1.0f);
}

__global__ void k_rsqrt(float* __restrict__ p, int n) {
  long long i = gtid();
  if (i < n) p[i] = rsqrtf(p[i]);
}

// ---------------- fp32 WMMA GEMM: C[M,N] = A[M,K] @ B[K,N] ----------------
// One wave computes a 32x16 output tile (two 16x16 WMMA accumulators sharing the
// B fragment). Main K loop is branch-free (EXEC stays all-1s); the K%4 tail uses
// clamped addresses + multiplicative masks (v_cndmask, no EXEC divergence).
// Fragment layouts per CDNA5 ISA:
//   A (16x4): lanes 0-15 rows M=0..15, v2f = {K, K+1}; lanes 16-31 = {K+2, K+3}.
//   B (4x16): lanes 0-15 col N, rows {K, K+1}; lanes 16-31 rows {K+2, K+3}.
//   C/D (16x16): vgpr r = row r (lanes 0-15) / row r+8 (lanes 16-31), col = lane&15.
__global__ void k_gemm_wmma(const float* __restrict__ A, const float* __restrict__ B,
                            float* __restrict__ C, int M, int K, int N) {
  int waveId = (int)(gtid() >> 5);
  int lane   = threadIdx.x & 31;
  int tilesN = N >> 4;
  int tiles  = (M >> 5) * tilesN;
  if (waveId >= tiles) return;              // wave-uniform exit
  int m0   = (waveId / tilesN) << 5;
  int n0   = (waveId % tilesN) << 4;
  int half = lane >> 4;                     // 0 or 1
  int l15  = lane & 15;

  const float* Arow0 = A + (long long)(m0 + l15) * K;
  const float* Arow1 = Arow0 + (long long)16 * K;
  const float* Bcol  = B + n0 + l15;

  v8f acc0 = {0.f, 0.f, 0.f, 0.f, 0.f, 0.f, 0.f, 0.f};
  v8f acc1 = {0.f, 0.f, 0.f, 0.f, 0.f, 0.f, 0.f, 0.f};

  int Kmain = K & ~3;
  int ka = half << 1;
  for (int k0 = 0; k0 < Kmain; k0 += 4, ka += 4) {
    v2f a0 = *(const v2f*)(Arow0 + ka);     // 8B-aligned: K and ka both even
    v2f a1 = *(const v2f*)(Arow1 + ka);
    v2f b;
    b.x = Bcol[(long long)ka * N];
    b.y = Bcol[(long long)(ka + 1) * N];
    acc0 = __builtin_amdgcn_wmma_f32_16x16x4_f32(false, a0, false, b, (short)0, acc0,
                                                 false, false);
    acc1 = __builtin_amdgcn_wmma_f32_16x16x4_f32(false, a1, false, b, (short)0, acc1,
                                                 false, false);
  }
  if (K & 3) {                              // tail (decoder K=66): clamp + mask
    float w0 = (ka     < K) ? 1.f : 0.f;
    float w1 = (ka + 1 < K) ? 1.f : 0.f;
    int kc0 = (ka     < K) ? ka     : K - 1;
    int kc1 = (ka + 1 < K) ? ka + 1 : K - 1;
    v2f a0, a1, b;
    a0.x = Arow0[kc0] * w0;  a0.y = Arow0[kc1] * w1;
    a1.x = Arow1[kc0] * w0;  a1.y = Arow1[kc1] * w1;
    b.x  = Bcol[(long long)kc0 * N] * w0;
    b.y  = Bcol[(long long)kc1 * N] * w1;
    acc0 = __builtin_amdgcn_wmma_f32_16x16x4_f32(false, a0, false, b, (short)0, acc0,
                                                 false, false);
    acc1 = __builtin_amdgcn_wmma_f32_16x16x4_f32(false, a1, false, b, (short)0, acc1,
                                                 false, false);
  }
#pragma unroll
  for (int r = 0; r < 8; ++r) {
    C[(long long)(m0 + r + (half << 3)) * N + n0 + l15]      = acc0[r];
    C[(long long)(m0 + 16 + r + (half << 3)) * N + n0 + l15] = acc1[r];
  }
}

// ---------------- GCN scatter: out[dst] += dinv[s]*dinv[d] * hw[src] ----------------
__global__ void k_scatter(const int* __restrict__ src, const int* __restrict__ dst,
                          const float* __restrict__ dinv, const float* __restrict__ hw,
                          float* __restrict__ out, int E, int F) {
  long long idx = gtid();
  long long total = (long long)E * F;
  if (idx >= total) return;
  int e = (int)(idx / F);
  int f = (int)(idx % F);
  int s = src[e], d = dst[e];
  atomicAdd(&out[(long long)d * F + f], dinv[s] * dinv[d] * hw[(long long)s * F + f]);
}

// acc = relu(acc + dinv[i]^2 * hw + bias)   (self-loop + bias + relu fused)
__global__ void k_finalize(float* __restrict__ acc, const float* __restrict__ hw,
                           const float* __restrict__ dinv, const float* __restrict__ bias,
                           int N, int F) {
  long long idx = gtid();
  if (idx >= (long long)N * F) return;
  int i = (int)(idx / F);
  int f = (int)(idx % F);
  float dd = dinv[i];
  acc[idx] = fmaxf(acc[idx] + dd * dd * hw[idx] + bias[f], 0.f);
}

// ---------------- global mean pool (LDS-staged partials) ----------------
__global__ void k_pool(const float* __restrict__ feat, const int* __restrict__ batch,
                       float* __restrict__ outg, float* cnt, int N) {
  __shared__ float s[NB * LATD];
  __shared__ float sc[NB];
  for (int t = threadIdx.x; t < NB * LATD; t += blockDim.x) s[t] = 0.f;
  if (threadIdx.x < NB) sc[threadIdx.x] = 0.f;
  __syncthreads();
  int f   = threadIdx.x & 63;
  int sub = threadIdx.x >> 6;                         // 0..3 (blockDim 256)
  int start = blockIdx.x * 512;
  int end   = start + 512 < N ? start + 512 : N;
  for (int i = start + sub; i < end; i += 4) {
    int b = batch[i];
    atomicAdd(&s[b * LATD + f], feat[(long long)i * LATD + f]);
    if (f == 0) atomicAdd(&sc[b], 1.0f);
  }
  __syncthreads();
  for (int t = threadIdx.x; t < NB * LATD; t += blockDim.x) atomicAdd(&outg[t], s[t]);
  if (cnt && threadIdx.x < NB) atomicAdd(&cnt[threadIdx.x], sc[threadIdx.x]);
}

__global__ void k_div(float* __restrict__ mu_g, float* __restrict__ lg_g,
                      const float* __restrict__ cnt) {
  int i = (int)gtid();
  if (i < NB * LATD) {
    float c = fmaxf(cnt[i >> 6], 1.0f);
    mu_g[i] /= c;
    lg_g[i] /= c;
  }
}

// ---------------- decoder input: Zp[r] = concat(mu_g[b], pos[j]) ----------------
__global__ void k_zp(const float* __restrict__ mu_g, const float* __restrict__ pos,
                     float* __restrict__ zp) {
  long long idx = gtid();
  if (idx >= (long long)ROWS * DEC_IN) return;
  int r = (int)(idx / DEC_IN);
  int c = (int)(idx % DEC_IN);
  int b = r >> 12;        // / 4096
  int j = r & 4095;
  zp[idx] = (c < LATD) ? mu_g[b * LATD + c] : pos[j * 2 + (c - LATD)];
}

// ---------------- LayerNorm(+bias in)+affine+ReLU, one wave per row of 128 ----------------
__global__ void k_ln_relu(const float* __restrict__ in, const float* __restrict__ bi,
                          const float* __restrict__ g, const float* __restrict__ be,
                          float* __restrict__ out, int rows) {
  int wave = (int)(gtid() >> 5);
  int lane = threadIdx.x & 31;
  if (wave >= rows) return;
  float v[4], s = 0.f, s2 = 0.f;
#pragma unroll
  for (int j = 0; j < 4; ++j) {
    int f = lane + 32 * j;
    v[j] = in[(long long)wave * DEC_H + f] + bi[f];
    s += v[j];
    s2 += v[j] * v[j];
  }
#pragma unroll
  for (int m = 16; m >= 1; m >>= 1) {
    s  += __shfl_xor(s, m, 32);
    s2 += __shfl_xor(s2, m, 32);
  }
  float mean = s * (1.f / DEC_H);
  float var  = s2 * (1.f / DEC_H) - mean * mean;
  float inv  = rsqrtf(var + 1e-5f);
#pragma unroll
  for (int j = 0; j < 4; ++j) {
    int f = lane + 32 * j;
    out[(long long)wave * DEC_H + f] = fmaxf((v[j] - mean) * inv * g[f] + be[f], 0.f);
  }
}

// ---------------- node heads: x = h + nm_bo; 8 narrow projections ----------------
__global__ void k_heads(const float* __restrict__ h, const float* __restrict__ bo,
                        const float* __restrict__ exw, const float* __restrict__ exb,
                        const float* __restrict__ fuw, const float* __restrict__ fub,
                        const float* __restrict__ alw, const float* __restrict__ alb,
                        const float* __restrict__ slw, const float* __restrict__ slb,
                        float* __restrict__ o_nodes, float* __restrict__ o_fuel,
                        float* __restrict__ o_alt, float* __restrict__ o_slope) {
  int r = (int)gtid();
  if (r >= ROWS) return;
  const float* hr = h + (long long)r * DEC_H;
  float a0 = 0, a1 = 0, a2 = 0, a3 = 0, a4 = 0, a5 = 0, a6 = 0, a7 = 0;
#pragma unroll 4
  for (int f = 0; f < DEC_H; ++f) {
    float x = hr[f] + bo[f];
    a0 += x * exw[f];
    a1 += x * fuw[f * 5 + 0];
    a2 += x * fuw[f * 5 + 1];
    a3 += x * fuw[f * 5 + 2];
    a4 += x * fuw[f * 5 + 3];
    a5 += x * fuw[f * 5 + 4];
    a6 += x * alw[f];
    a7 += x * slw[f];
  }
  o_nodes[r] = a0 + exb[0];
  o_fuel[r * 5 + 0] = a1 + fub[0];
  o_fuel[r * 5 + 1] = a2 + fub[1];
  o_fuel[r * 5 + 2] = a3 + fub[2];
  o_fuel[r * 5 + 3] = a4 + fub[3];
  o_fuel[r * 5 + 4] = a5 + fub[4];
  o_alt[r]   = a6 + alb[0];
  o_slope[r] = a7 + slb[0];
}

// ---------------- edge mlp output layer 128 -> 8 ----------------
__global__ void k_edgeout(const float* __restrict__ h, const float* __restrict__ wo,
                          const float* __restrict__ bo, float* __restrict__ out) {
  int r = (int)gtid();
  if (r >= ROWS) return;
  const float* hr = h + (long long)r * DEC_H;
  float a[8] = {0, 0, 0, 0, 0, 0, 0, 0};
#pragma unroll 4
  for (int f = 0; f < DEC_H; ++f) {
    float x = hr[f];
#pragma unroll
    for (int j = 0; j < 8; ++j) a[j] += x * wo[f * 8 + j];
  }
#pragma unroll
  for (int j = 0; j < 8; ++j) out[r * 8 + j] = a[j] + bo[j];
}

// ---------------- host side ----------------
static inline int cdiv(long long a, int b) { return (int)((a + b - 1) / b); }

extern "C" void kernel_launch(void* const* d_in, const int* in_sizes, int n_in,
                              void* d_out, int out_size, void* d_ws, size_t ws_size,
                              hipStream_t stream) {
  (void)in_sizes; (void)n_in; (void)out_size; (void)ws_size;

  const float* x     = (const float*)d_in[0];
  const int*   ei    = (const int*)d_in[1];
  const int*   batch = (const int*)d_in[2];
  const int*   src   = ei;
  const int*   dst   = ei + NE;
  const float *w1m = (const float*)d_in[3],  *b1m = (const float*)d_in[4];
  const float *w2m = (const float*)d_in[5],  *b2m = (const float*)d_in[6];
  const float *w3m = (const float*)d_in[7],  *b3m = (const float*)d_in[8];
  const float *w1l = (const float*)d_in[9],  *b1l = (const float*)d_in[10];
  const float *w2l = (const float*)d_in[11], *b2l = (const float*)d_in[12];
  const float *w3l = (const float*)d_in[13], *b3l = (const float*)d_in[14];
  const float *nm_wi = (const float*)d_in[15], *nm_bi = (const float*)d_in[16];
  const float *nm_g  = (const float*)d_in[17], *nm_be = (const float*)d_in[18];
  const float *nm_wo = (const float*)d_in[19], *nm_bo = (const float*)d_in[20];
  const float *ex_w = (const float*)d_in[21], *ex_b = (const float*)d_in[22];
  const float *fu_w = (const float*)d_in[23], *fu_b = (const float*)d_in[24];
  const float *al_w = (const float*)d_in[25], *al_b = (const float*)d_in[26];
  const float *sl_w = (const float*)d_in[27], *sl_b = (const float*)d_in[28];
  const float *em_wi = (const float*)d_in[29], *em_bi = (const float*)d_in[30];
  const float *em_g  = (const float*)d_in[31], *em_be = (const float*)d_in[32];
  const float *em_wo = (const float*)d_in[33], *em_bo = (const float*)d_in[34];
  const float *pos_all = (const float*)d_in[35];

  float* out = (float*)d_out;
  // output layout (flat, return order)
  float* o_nodes = out;                 // 65536
  float* o_edges = out + 65536;         // 65536*8
  float* o_fuel  = out + 589824;        // 65536*5
  float* o_alt   = out + 917504;        // 65536
  float* o_slope = out + 983040;        // 65536
  float* o_mug   = out + 1048576;       // 1024
  float* o_lgg   = out + 1049600;       // 1024

  // workspace layout (floats): dinv | cnt | bufA(128w) | bufB(128w) | bufC(64w)
  float* W    = (float*)d_ws;
  float* dinv = W;                        // 100000
  float* cnt  = W + 100352;               // 16
  float* bufA = W + 131072;               // 12,800,000  (100000*128)
  float* bufB = bufA + 12800000;          // 12,800,000
  float* bufC = bufB + 12800000;          // 6,400,000   (100000*64)

  const int TB = 256;
  auto fill = [&](float* p, float v, long long n) {
    k_fill<<<cdiv(n, TB), TB, 0, stream>>>(p, v, n);
  };
  auto gemm = [&](const float* A, const float* B, float* C, int M, int K, int N) {
    long long thr = (long long)(M / 32) * (N / 16) * 32;   // one wave per 32x16 tile
    k_gemm_wmma<<<cdiv(thr, TB), TB, 0, stream>>>(A, B, C, M, K, N);
  };

  // ---- degrees & symmetric norm (self-loops give deg >= 1) ----
  fill(dinv, 1.0f, NN);
  k_deg<<<cdiv(NE, TB), TB, 0, stream>>>(dst, dinv, NE);
  k_rsqrt<<<cdiv(NN, TB), TB, 0, stream>>>(dinv, NN);

  // ---- one GCN layer: acc = relu(scatter(norm * (h@W)) + selfloop + bias) ----
  auto layer = [&](const float* hin, const float* wt, const float* bt,
                   float* hw, float* acc, int K, int F) {
    gemm(hin, wt, hw, NN, K, F);
    fill(acc, 0.f, (long long)NN * F);
    k_scatter<<<cdiv((long long)NE * F, TB), TB, 0, stream>>>(src, dst, dinv, hw, acc, NE, F);
    k_finalize<<<cdiv((long long)NN * F, TB), TB, 0, stream>>>(acc, hw, dinv, bt, NN, F);
  };

  auto branch = [&](const float* w1, const float* b1, const float* w2, const float* b2,
                    const float* w3, const float* b3, float* outg, float* cntp) {
    layer(x,    w1, b1, bufA, bufC, 32, 64);    // h1 -> bufC (64w)
    layer(bufC, w2, b2, bufA, bufB, 64, 128);   // h2 -> bufB (128w)
    layer(bufB, w3, b3, bufC, bufA, 128, 64);   // h3 -> bufA (64w)
    fill(outg, 0.f, NB * LATD);
    if (cntp) fill(cntp, 0.f, NB);
    k_pool<<<cdiv(NN, 512), TB, 0, stream>>>(bufA, batch, outg, cntp, NN);
  };

  branch(w1m, b1m, w2m, b2m, w3m, b3m, o_mug, cnt);      // mu branch (+counts)
  branch(w1l, b1l, w2l, b2l, w3l, b3l, o_lgg, nullptr);  // logvar branch
  k_div<<<cdiv(NB * LATD, TB), TB, 0, stream>>>(o_mug, o_lgg, cnt);

  // ---- decoder ----
  k_zp<<<cdiv((long long)ROWS * DEC_IN, TB), TB, 0, stream>>>(o_mug, pos_all, bufC);

  // node_mlp: Zp @ nm_wi -> LN+ReLU -> @ nm_wo -> heads
  gemm(bufC, nm_wi, bufA, ROWS, DEC_IN, DEC_H);
  k_ln_relu<<<cdiv((long long)ROWS * 32, TB), TB, 0, stream>>>(bufA, nm_bi, nm_g, nm_be,
                                                               bufB, ROWS);
  gemm(bufB, nm_wo, bufA, ROWS, DEC_H, DEC_H);
  k_heads<<<cdiv(ROWS, TB), TB, 0, stream>>>(bufA, nm_bo, ex_w, ex_b, fu_w, fu_b,
                                             al_w, al_b, sl_w, sl_b,
                                             o_nodes, o_fuel, o_alt, o_slope);

  // edge_mlp: Zp @ em_wi -> LN+ReLU -> @ em_wo (+bo)
  gemm(bufC, em_wi, bufA, ROWS, DEC_IN, DEC_H);
  k_ln_relu<<<cdiv((long long)ROWS * 32, TB), TB, 0, stream>>>(bufA, em_bi, em_g, em_be,
                                                               bufB, ROWS);
  k_edgeout<<<cdiv(ROWS, TB), TB, 0, stream>>>(bufB, em_wo, em_bo, o_edges);
}